// GraphConvolution_27367531610429
// MI455X (gfx1250) — compile-verified
//
#include <hip/hip_runtime.h>
#include <cmath>

typedef __attribute__((ext_vector_type(2))) float v2f;
typedef __attribute__((ext_vector_type(8))) float v8f;

#define D 128
#define WSTRIDE 132   // padded LDS row stride (floats): conflict-free b64 reads

// ---------------------------------------------------------------------------
// Kernel 1: support = input @ W^T + b  via V_WMMA_F32_16X16X4_F32.
// W staged in LDS once per workgroup (64KB of the 320KB WGP LDS).
// One wave computes a 16-row strip across all 128 output columns.
// ---------------------------------------------------------------------------
__global__ void __launch_bounds__(256)
gcn_gemm_wmma(const float* __restrict__ in, const float* __restrict__ W,
              const float* __restrict__ bias, float* __restrict__ sup,
              int nTiles) {
  __shared__ float Wlds[D * WSTRIDE];

  // Cooperative staging: 256 threads copy 128x128 floats, padded rows.
  for (int i = threadIdx.x; i < (D * D) / 4; i += 256) {
    const int row = i / (D / 4);
    const int c4  = i % (D / 4);
    const float4 w = *(const float4*)(W + (long)row * D + c4 * 4);
    float* dst = &Wlds[row * WSTRIDE + c4 * 4];
    dst[0] = w.x; dst[1] = w.y; dst[2] = w.z; dst[3] = w.w;
  }
  __syncthreads();

  const int wid  = (blockIdx.x * (blockDim.x >> 5)) + (threadIdx.x >> 5);
  const int lane = threadIdx.x & 31;
  if (wid >= nTiles) return;              // wave-uniform: EXEC stays all-ones

  const int r     = lane & 15;            // M index (A) / N index (B,C)
  const int khalf = lane >> 4;            // 0: K={0,1}, 1: K={2,3}
  const long rowA = (long)wid * 16 + r;

  v8f acc[8];
  #pragma unroll
  for (int t = 0; t < 8; ++t) acc[t] = (v8f){0,0,0,0,0,0,0,0};

  for (int k = 0; k < D; k += 4) {
    const int kk = k + khalf * 2;
    const v2f a = *(const v2f*)(in + rowA * D + kk);
    #pragma unroll
    for (int t = 0; t < 8; ++t) {
      const v2f bf = *(const v2f*)(&Wlds[(t * 16 + r) * WSTRIDE + kk]);
      acc[t] = __builtin_amdgcn_wmma_f32_16x16x4_f32(
          /*neg_a=*/false, a, /*neg_b=*/false, bf,
          /*c_mod=*/(short)0, acc[t], /*reuse_a=*/false, /*reuse_b=*/false);
    }
  }

  #pragma unroll
  for (int t = 0; t < 8; ++t) {
    const float bv = bias[t * 16 + r];
    #pragma unroll
    for (int j = 0; j < 8; ++j) {
      const int m = j + khalf * 8;        // C/D layout: vgpr j, lanes>=16 -> M=j+8
      sup[((long)wid * 16 + m) * D + t * 16 + r] = acc[t][j] + bv;
    }
  }
}

// Scalar fallback for a ragged row tail (N % 16) — not taken for N=50000.
__global__ void gcn_gemm_tail(const float* __restrict__ in, const float* __restrict__ W,
                              const float* __restrict__ bias, float* __restrict__ sup,
                              int nStart, int N) {
  const int idx = blockIdx.x * blockDim.x + threadIdx.x;
  const int n = nStart + idx / D;
  const int o = idx % D;
  if (n >= N) return;
  float s = bias[o];
  for (int d = 0; d < D; ++d) s += in[(long)n * D + d] * W[(long)o * D + d];
  sup[(long)n * D + o] = s;
}

// ---------------------------------------------------------------------------
// Kernel 2: out[n] = tanh( sum_{e: row[e]==n} val[e] * support[col[e]] )
// adj_row is sorted -> one wave per destination node; binary-search the edge
// range, gather support rows (L2-resident, 25.6MB < 192MB L2), accumulate in
// registers, write tanh directly. No atomics, no zero-init, no extra pass.
// ---------------------------------------------------------------------------
__global__ void __launch_bounds__(256)
gcn_aggregate(const int* __restrict__ rows, const int* __restrict__ cols,
              const float* __restrict__ vals, const float* __restrict__ sup,
              float* __restrict__ out, int N, int E) {
  const int node = (blockIdx.x * (blockDim.x >> 5)) + (threadIdx.x >> 5);
  const int lane = threadIdx.x & 31;
  if (node >= N) return;

  // lower_bound(rows, node) and lower_bound(rows, node+1): wave-uniform.
  int lo = 0, hi = E;
  while (lo < hi) {
    const int mid = (lo + hi) >> 1;
    if (rows[mid] < node) lo = mid + 1; else hi = mid;
  }
  const int start = lo;
  hi = E;
  while (lo < hi) {
    const int mid = (lo + hi) >> 1;
    if (rows[mid] <= node) lo = mid + 1; else hi = mid;
  }
  const int end = lo;

  const int fo = lane * 4;                 // lane owns features [4l, 4l+4)
  float4 acc = make_float4(0.f, 0.f, 0.f, 0.f);

  for (int base = start; base < end; base += 32) {
    const int cnt = min(32, end - base);
    const int ld  = base + min(lane, cnt - 1);   // coalesced metadata load
    const int   c32 = cols[ld];
    const float v32 = vals[ld];
    for (int i = 0; i < cnt; ++i) {
      const int   cc = __shfl(c32, i);
      const float vv = __shfl(v32, i);
      const float4 s = *(const float4*)(sup + (long)cc * D + fo);
      acc.x += s.x * vv;
      acc.y += s.y * vv;
      acc.z += s.z * vv;
      acc.w += s.w * vv;
    }
  }

  const float4 o = make_float4(tanhf(acc.x), tanhf(acc.y),
                               tanhf(acc.z), tanhf(acc.w));
  *(float4*)(out + (long)node * D + fo) = o;   // empty rows -> tanh(0) = 0
}

// ---------------------------------------------------------------------------
extern "C" void kernel_launch(void* const* d_in, const int* in_sizes, int n_in,
                              void* d_out, int out_size, void* d_ws, size_t ws_size,
                              hipStream_t stream) {
  const float* input = (const float*)d_in[0];
  const float* W     = (const float*)d_in[1];
  const float* bias  = (const float*)d_in[2];
  const int*   arow  = (const int*)d_in[3];
  const int*   acol  = (const int*)d_in[4];
  const float* aval  = (const float*)d_in[5];
  float* out = (float*)d_out;
  float* sup = (float*)d_ws;           // N*D floats of scratch (25.6 MB)

  const int N = in_sizes[0] / D;
  const int E = in_sizes[3];

  // 1) dense GEMM via f32 WMMA with LDS-staged W.
  const int nTiles = N / 16;
  if (nTiles > 0) {
    const int wavesPerBlk = 8;                       // 256 threads
    gcn_gemm_wmma<<<(nTiles + wavesPerBlk - 1) / wavesPerBlk, 256, 0, stream>>>(
        input, W, bias, sup, nTiles);
  }
  const int rem = N - nTiles * 16;
  if (rem > 0) {
    gcn_gemm_tail<<<(rem * D + 255) / 256, 256, 0, stream>>>(input, W, bias, sup,
                                                             nTiles * 16, N);
  }

  // 2) fused sorted-COO aggregation + tanh, one wave per node, no atomics.
  gcn_aggregate<<<(N + 7) / 8, 256, 0, stream>>>(arow, acol, aval, sup, out, N, E);
}